// GATNet_68496138436962
// MI455X (gfx1250) — compile-verified
//
#include <hip/hip_runtime.h>
#include <hip/hip_bf16.h>
#include <math.h>

// ---------------------------------------------------------------------------
// GATNet forward for MI455X (gfx1250, wave32).
// Only the node path matters (e_feat is deleted in the reference):
//   hh0 = h @ W_emb + b_emb
//   hh1 = GAT(hh0; fc1, al1, ar1, g1, b1, heads=8, dim=32, residual)
//   out = GAT(hh1; fc2, al2, ar2, g2, b2, heads=1, dim=32, no residual)
// GEMMs run on V_WMMA_F32_16X16X4_F32 (exact fp32, shapes tile perfectly).
// ---------------------------------------------------------------------------

typedef __attribute__((ext_vector_type(2))) float v2f;
typedef __attribute__((ext_vector_type(8))) float v8f;

#define NNODES 10000
#define NEDGES 120000

// ---- monotonic float<->uint encoding for atomicMax-based segment max ------
__device__ __forceinline__ unsigned encF(float f) {
  unsigned b = __float_as_uint(f);
  return (b & 0x80000000u) ? ~b : (b | 0x80000000u);
}
__device__ __forceinline__ float decF(unsigned u) {
  return __uint_as_float((u & 0x80000000u) ? (u ^ 0x80000000u) : ~u);
}

__device__ __forceinline__ float atomAddF(float* p, float v) {
  return unsafeAtomicAdd(p, v);   // -> global_atomic_add_f32
}

// ---------------------------------------------------------------------------
// WMMA f32 GEMM: C[M,Nc] = A[M,K] @ B[K,Nc] (+ bias broadcast over rows).
// One wave computes one 16x16 tile; K-loop in steps of 4.
// Requires M%16==0, K%4==0, Nc%16==0 (holds for all three GEMMs here).
// ---------------------------------------------------------------------------
__global__ void gemm_wmma_f32(const float* __restrict__ A,
                              const float* __restrict__ B,
                              const float* __restrict__ bias,
                              float* __restrict__ C,
                              int M, int K, int Nc) {
  const int lane  = threadIdx.x & 31;
  const int wave  = (blockIdx.x * (blockDim.x >> 5)) + (threadIdx.x >> 5);
  const int tilesN = Nc >> 4;
  const int totalTiles = (M >> 4) * tilesN;
  if (wave >= totalTiles) return;            // wave-uniform: EXEC stays all-1s

  const int tileM = wave / tilesN;
  const int tileN = wave % tilesN;
  const int hi = lane >> 4;                  // 0: K pair {0,1}, 1: K pair {2,3}
  const int lo = lane & 15;

  const float* Arow = A + (size_t)(tileM * 16 + lo) * K;   // lane = M row
  const float* Bcol = B + (size_t)(tileN * 16 + lo);       // lane = N col

  v8f c = {};
  for (int k = 0; k < K; k += 4) {
    const int ka = k + 2 * hi;
    v2f a, b;
    a.x = Arow[ka];
    a.y = Arow[ka + 1];
    b.x = Bcol[(size_t)ka * Nc];
    b.y = Bcol[(size_t)(ka + 1) * Nc];
    c = __builtin_amdgcn_wmma_f32_16x16x4_f32(
            /*neg_a=*/false, a, /*neg_b=*/false, b,
            /*c_mod=*/(short)0, c, /*reuse_a=*/false, /*reuse_b=*/false);
  }

  const int col = tileN * 16 + lo;
  const float bv = bias ? bias[col] : 0.0f;
#pragma unroll
  for (int r = 0; r < 8; ++r) {
    const int row = tileM * 16 + r + 8 * hi;   // C layout: VGPR r, halves M/M+8
    C[(size_t)row * Nc + col] = c[r] + bv;
  }
}

// ---------------------------------------------------------------------------
// el[n,h] = sum_d feat[n,h,d]*al[h,d] ; er likewise
// ---------------------------------------------------------------------------
__global__ void attn_coef(const float* __restrict__ feat,
                          const float* __restrict__ al,
                          const float* __restrict__ ar,
                          float* __restrict__ el, float* __restrict__ er,
                          int N, int heads, int dim) {
  const int i = blockIdx.x * blockDim.x + threadIdx.x;   // n*heads + h
  if (i >= N * heads) return;
  const int h = i % heads;
  const float* f = feat + (size_t)i * dim;
  float sl = 0.f, sr = 0.f;
  for (int d = 0; d < dim; ++d) {
    const float v = f[d];
    sl += v * al[h * dim + d];
    sr += v * ar[h * dim + d];
  }
  el[i] = sl;
  er[i] = sr;
}

__device__ __forceinline__ float leaky02(float x) {
  return x > 0.f ? x : 0.2f * x;
}

// ---------------------------------------------------------------------------
// Pass 1 over edges: per-(dst,head) running max of leaky_relu(el[src]+er[dst])
// ---------------------------------------------------------------------------
__global__ void edge_max(const int* __restrict__ src, const int* __restrict__ dst,
                         const float* __restrict__ el, const float* __restrict__ er,
                         unsigned* __restrict__ menc, int E, int heads) {
  const int i = blockIdx.x * blockDim.x + threadIdx.x;
  if (i >= E * heads) return;
  const int e = i / heads, h = i % heads;
  const float x = leaky02(el[src[e] * heads + h] + er[dst[e] * heads + h]);
  atomicMax(&menc[dst[e] * heads + h], encF(x));
}

// ---------------------------------------------------------------------------
// Pass 2: ex = exp(logit - m[dst]); s[dst,h] += ex;
//         num[dst,h,:] += ex * feat[src,h,:]
// ---------------------------------------------------------------------------
__global__ void edge_accum(const int* __restrict__ src, const int* __restrict__ dst,
                           const float* __restrict__ el, const float* __restrict__ er,
                           const unsigned* __restrict__ menc,
                           const float* __restrict__ feat,
                           float* __restrict__ s, float* __restrict__ num,
                           int E, int heads, int dim) {
  const int i = blockIdx.x * blockDim.x + threadIdx.x;
  if (i >= E * heads) return;
  const int e = i / heads, h = i % heads;
  const int sn = src[e], dn = dst[e];
  const float x  = leaky02(el[sn * heads + h] + er[dn * heads + h]);
  const float ex = expf(x - decF(menc[dn * heads + h]));
  atomAddF(&s[dn * heads + h], ex);
  const float* f = feat + ((size_t)sn * heads + h) * dim;
  float* o = num + ((size_t)dn * heads + h) * dim;
  for (int d = 0; d < dim; ++d) atomAddF(&o[d], ex * f[d]);
}

// rst = num / (s + 1e-16), in place
__global__ void rst_div(float* __restrict__ num, const float* __restrict__ s,
                        int N, int heads, int dim) {
  const int i = blockIdx.x * blockDim.x + threadIdx.x;
  if (i >= N * heads * dim) return;
  num[i] = num[i] / (s[i / dim] + 1e-16f);
}

// ---------------------------------------------------------------------------
// BatchNorm stats per channel (biased variance, as jnp.var ddof=0)
// one block per channel c
// ---------------------------------------------------------------------------
__global__ void bn_stats(const float* __restrict__ x,
                         float* __restrict__ mean, float* __restrict__ var,
                         int N, int C) {
  const int c = blockIdx.x;
  float s = 0.f, s2 = 0.f;
  for (int n = threadIdx.x; n < N; n += blockDim.x) {
    const float v = x[(size_t)n * C + c];
    s += v;
    s2 += v * v;
  }
  __shared__ float sh[256], sh2[256];
  sh[threadIdx.x] = s;
  sh2[threadIdx.x] = s2;
  __syncthreads();
  for (int off = blockDim.x >> 1; off > 0; off >>= 1) {
    if (threadIdx.x < off) {
      sh[threadIdx.x]  += sh[threadIdx.x + off];
      sh2[threadIdx.x] += sh2[threadIdx.x + off];
    }
    __syncthreads();
  }
  if (threadIdx.x == 0) {
    const float mu = sh[0] / (float)N;
    mean[c] = mu;
    var[c]  = sh2[0] / (float)N - mu * mu;
  }
}

// y = [resid +] elu(g*(x-mean)*rsqrt(var+eps)+b)
__global__ void bn_apply(const float* __restrict__ x,
                         const float* __restrict__ mean, const float* __restrict__ var,
                         const float* __restrict__ g, const float* __restrict__ b,
                         const float* __restrict__ resid,
                         float* __restrict__ y, int N, int C) {
  const int i = blockIdx.x * blockDim.x + threadIdx.x;
  if (i >= N * C) return;
  const int c = i % C;
  float v = g[c] * (x[i] - mean[c]) * rsqrtf(var[c] + 1e-5f) + b[c];
  v = v > 0.f ? v : (expf(v) - 1.0f);   // ELU(alpha=1)
  if (resid) v += resid[i];
  y[i] = v;
}

// ---------------------------------------------------------------------------
extern "C" void kernel_launch(void* const* d_in, const int* in_sizes, int n_in,
                              void* d_out, int out_size, void* d_ws, size_t ws_size,
                              hipStream_t stream) {
  const float* h     = (const float*)d_in[0];
  const int*   src   = (const int*)  d_in[2];
  const int*   dst   = (const int*)  d_in[3];
  const float* W_emb = (const float*)d_in[4];
  const float* b_emb = (const float*)d_in[5];
  const float* fc1   = (const float*)d_in[18];
  const float* al1   = (const float*)d_in[19];
  const float* ar1   = (const float*)d_in[20];
  const float* g1    = (const float*)d_in[21];
  const float* b1    = (const float*)d_in[22];
  const float* fc2   = (const float*)d_in[23];
  const float* al2   = (const float*)d_in[24];
  const float* ar2   = (const float*)d_in[25];
  const float* g2    = (const float*)d_in[26];
  const float* b2    = (const float*)d_in[27];
  float* out = (float*)d_out;

  const int N = NNODES, E = NEDGES;
  const int HD = 256, HEADS = 8, DIM = 32;

  // ---- workspace layout (floats) ----
  float* ws = (float*)d_ws;
  size_t off = 0;
  float* hh0   = ws + off; off += (size_t)N * HD;   // 2,560,000
  float* feat1 = ws + off; off += (size_t)N * HD;
  float* hh1   = ws + off; off += (size_t)N * HD;
  float* el1   = ws + off; off += (size_t)N * HEADS;
  float* er1   = ws + off; off += (size_t)N * HEADS;
  float* feat2 = ws + off; off += (size_t)N * DIM;
  float* el2   = ws + off; off += (size_t)N;
  float* er2   = ws + off; off += (size_t)N;
  // zero-initialized block (one contiguous memset)
  float*    zero_base = ws + off;
  float*    num1  = ws + off; off += (size_t)N * HD;
  float*    s1    = ws + off; off += (size_t)N * HEADS;
  unsigned* m1enc = (unsigned*)(ws + off); off += (size_t)N * HEADS;
  float*    num2  = ws + off; off += (size_t)N * DIM;
  float*    s2    = ws + off; off += (size_t)N;
  unsigned* m2enc = (unsigned*)(ws + off); off += (size_t)N;
  const size_t zero_bytes = ((ws + off) - zero_base) * sizeof(float);
  // stats
  float* mean1 = ws + off; off += HD;
  float* var1  = ws + off; off += HD;
  float* mean2 = ws + off; off += DIM;
  float* var2  = ws + off; off += DIM;

  hipMemsetAsync(zero_base, 0, zero_bytes, stream);

  const int TB = 256;                 // 8 waves / block
  auto cdiv = [](int a, int b) { return (a + b - 1) / b; };

  // 1) hh0 = h @ W_emb + b_emb          [10000,128]x[128,256]
  {
    const int tiles = (N / 16) * (HD / 16);
    gemm_wmma_f32<<<cdiv(tiles, 8), TB, 0, stream>>>(h, W_emb, b_emb, hh0, N, 128, HD);
  }

  // ---- GAT layer 1 (8 heads x 32, residual) ----
  {
    const int tiles = (N / 16) * (HD / 16);
    gemm_wmma_f32<<<cdiv(tiles, 8), TB, 0, stream>>>(hh0, fc1, nullptr, feat1, N, HD, HD);
  }
  attn_coef<<<cdiv(N * HEADS, TB), TB, 0, stream>>>(feat1, al1, ar1, el1, er1, N, HEADS, DIM);
  edge_max<<<cdiv(E * HEADS, TB), TB, 0, stream>>>(src, dst, el1, er1, m1enc, E, HEADS);
  edge_accum<<<cdiv(E * HEADS, TB), TB, 0, stream>>>(src, dst, el1, er1, m1enc, feat1,
                                                     s1, num1, E, HEADS, DIM);
  rst_div<<<cdiv(N * HD, TB), TB, 0, stream>>>(num1, s1, N, HEADS, DIM);
  bn_stats<<<HD, TB, 0, stream>>>(num1, mean1, var1, N, HD);
  bn_apply<<<cdiv(N * HD, TB), TB, 0, stream>>>(num1, mean1, var1, g1, b1, hh0, hh1, N, HD);

  // ---- GAT layer 2 (1 head x 32, no residual) ----
  {
    const int tiles = (N / 16) * (DIM / 16);
    gemm_wmma_f32<<<cdiv(tiles, 8), TB, 0, stream>>>(hh1, fc2, nullptr, feat2, N, HD, DIM);
  }
  attn_coef<<<cdiv(N, TB), TB, 0, stream>>>(feat2, al2, ar2, el2, er2, N, 1, DIM);
  edge_max<<<cdiv(E, TB), TB, 0, stream>>>(src, dst, el2, er2, m2enc, E, 1);
  edge_accum<<<cdiv(E, TB), TB, 0, stream>>>(src, dst, el2, er2, m2enc, feat2,
                                             s2, num2, E, 1, DIM);
  rst_div<<<cdiv(N * DIM, TB), TB, 0, stream>>>(num2, s2, N, 1, DIM);
  bn_stats<<<DIM, TB, 0, stream>>>(num2, mean2, var2, N, DIM);
  bn_apply<<<cdiv(N * DIM, TB), TB, 0, stream>>>(num2, mean2, var2, g2, b2, nullptr,
                                                 out, N, DIM);
  (void)in_sizes; (void)n_in; (void)out_size; (void)ws_size;
}